// LiteMLA_67001489817783
// MI455X (gfx1250) — compile-verified
//
#include <hip/hip_runtime.h>

// ---------------------------------------------------------------------------
// LiteMLA fused kernel for MI455X (gfx1250, wave32).
//   Pass 1: qkv(1x1 conv)+relu -> dwconv{1,3,5}+relu -> qk = Q K^T summed
//           over T via v_wmma_f32_16x16x4_f32 (all 4 heads batched into one
//           16x16 accumulator), atomically reduced into d_ws.
//   Pass 2: recompute v-channels, out = qk(block-diag) @ V via WMMA,
//           8x48 projection + relu, coalesced (B,T,8) store.
// ---------------------------------------------------------------------------

#define B_    64
#define T_    16384
#define C_    8
#define TD3   48
#define CHUNK 128
#define HALO  2
#define QKV_T (CHUNK + 2 * HALO)   // 132

typedef __attribute__((ext_vector_type(2))) float v2f;
typedef __attribute__((ext_vector_type(8))) float v8f;

// ---------------------------------------------------------------------------
__global__ void zero_ws_kernel(float* p, int n) {
    int i = blockIdx.x * 256 + threadIdx.x;
    if (i < n) p[i] = 0.0f;
}

// ---------------------------------------------------------------------------
// Stage A helper: compute relu(W_qkv x + b) for the chunk + halo into LDS,
// laid out channel-major sh_qkv[ch][QKV_T]. Out-of-range tokens -> 0
// (the depthwise conv zero-pads its *input*, which is this qkv tensor).
__device__ __forceinline__ void stage_qkv(
    const float* __restrict__ x, const float* __restrict__ w_qkv,
    const float* __restrict__ b_qkv, float* __restrict__ sh_qkv,
    int b, int t0, int tid)
{
    for (int tl = tid; tl < QKV_T; tl += 256) {
        int t = t0 + tl - HALO;
        bool inb = (t >= 0) && (t < T_);
        float xv[C_];
        if (inb) {
            const float* xp = x + ((size_t)b * T_ + t) * C_;
            __builtin_prefetch(xp + 256 * C_, 0, 0);   // global_prefetch_b8
#pragma unroll
            for (int c = 0; c < C_; ++c) xv[c] = xp[c];
        }
#pragma unroll 4
        for (int o = 0; o < TD3; ++o) {
            float acc = 0.0f;
            if (inb) {
                acc = b_qkv[o];
#pragma unroll
                for (int c = 0; c < C_; ++c) acc += w_qkv[o * C_ + c] * xv[c];
                acc = fmaxf(acc, 0.0f);
            }
            sh_qkv[o * QKV_T + tl] = acc;
        }
    }
}

// Depthwise conv (+bias+relu) for one output element, reading staged qkv.
__device__ __forceinline__ float dwconv_one(
    const float* __restrict__ sh_qkv, const float* __restrict__ w,
    const float* __restrict__ bb, int ch, int tl, int k)
{
    float acc = bb[ch];
    const float* q = &sh_qkv[ch * QKV_T + (tl + HALO)];
#pragma unroll 5
    for (int j = 0; j < 5; ++j) {
        if (j < k) acc += w[ch * k + j] * q[j - (k >> 1)];
    }
    return fmaxf(acc, 0.0f);
}

// ---------------------------------------------------------------------------
// Pass 1: accumulate qk[b][branch][16][16] = sum_t Q K^T
__global__ void __launch_bounds__(256)
litemla_pass1(const float* __restrict__ x,
              const float* __restrict__ w_qkv, const float* __restrict__ b_qkv,
              const float* __restrict__ w1, const float* __restrict__ bb1,
              const float* __restrict__ w3, const float* __restrict__ bb3,
              const float* __restrict__ w5, const float* __restrict__ bb5,
              float* __restrict__ qk_ws)
{
    extern __shared__ float smem[];
    float* sh_qkv = smem;                        // [48][QKV_T]
    float* sh_s   = smem + TD3 * QKV_T;          // [3][32][CHUNK] (q=0..15,k=16..31)

    const int b   = blockIdx.y;
    const int t0  = blockIdx.x * CHUNK;
    const int tid = threadIdx.x;

    stage_qkv(x, w_qkv, b_qkv, sh_qkv, b, t0, tid);
    __syncthreads();

    // Stage B: s = relu(dwconv(qkv)) for q,k channels of all 3 branches.
    for (int idx = tid; idx < 3 * 32 * CHUNK; idx += 256) {
        int tl = idx & (CHUNK - 1);
        int ch = (idx / CHUNK) & 31;        // channel 0..31 (q then k)
        int br = idx / (CHUNK * 32);        // branch 0..2
        const float *w, *bb; int k;
        if      (br == 0) { w = w1; bb = bb1; k = 1; }
        else if (br == 1) { w = w3; bb = bb3; k = 3; }
        else              { w = w5; bb = bb5; k = 5; }
        sh_s[(br * 32 + ch) * CHUNK + tl] = dwconv_one(sh_qkv, w, bb, ch, tl, k);
    }
    __syncthreads();

    // Stage C: WMMA reduction. 6 active waves: wave = branch*2 + half.
    const int wave = tid >> 5;
    const int lane = tid & 31;
    if (wave < 6) {
        const int br   = wave >> 1;
        const int half = wave & 1;
        const int grp  = lane >> 4;     // 0: K=0,1  1: K=2,3
        const int r    = lane & 15;     // matrix row (A: M, B: N)
        const float* shQ = &sh_s[(br * 32 +  0) * CHUNK];
        const float* shK = &sh_s[(br * 32 + 16) * CHUNK];

        v8f acc = {};
        const int tbeg = half * (CHUNK / 2);
        const int tend = tbeg + (CHUNK / 2);
#pragma unroll 4
        for (int tau = tbeg; tau < tend; tau += 4) {
            const int off = tau + 2 * grp;
            // A[M=r][kk] = Q[r][tau+kk]; lane holds kk = 2*grp, 2*grp+1
            v2f a; a.x = shQ[r * CHUNK + off]; a.y = shQ[r * CHUNK + off + 1];
            // B[kk][N=r] = K[r][tau+kk]
            v2f bm; bm.x = shK[r * CHUNK + off]; bm.y = shK[r * CHUNK + off + 1];
            acc = __builtin_amdgcn_wmma_f32_16x16x4_f32(
                false, a, false, bm, (short)0, acc, false, false);
        }

        float* dst = qk_ws + ((size_t)(b * 3 + br)) * 256;
#pragma unroll
        for (int i = 0; i < 8; ++i) {
            int M = i + 8 * grp;          // C layout: VGPR i -> M = i + 8*(lane>=16)
            atomicAdd(&dst[M * 16 + r], acc[i]);
        }
    }
}

// ---------------------------------------------------------------------------
// Pass 2: out = (block-diag qk) @ V, concat, 8x48 projection + relu.
__global__ void __launch_bounds__(256)
litemla_pass2(const float* __restrict__ x,
              const float* __restrict__ w_qkv, const float* __restrict__ b_qkv,
              const float* __restrict__ w1, const float* __restrict__ bb1,
              const float* __restrict__ w3, const float* __restrict__ bb3,
              const float* __restrict__ w5, const float* __restrict__ bb5,
              const float* __restrict__ w_proj, const float* __restrict__ b_proj,
              const float* __restrict__ qk_ws, float* __restrict__ out)
{
    extern __shared__ float smem[];
    float* sh_qkv = smem;                                  // [48][QKV_T]
    float* sh_v   = smem + TD3 * QKV_T;                    // [3][16][CHUNK]
    float* sh_out = smem + TD3 * QKV_T + 3 * 16 * CHUNK;   // [48][CHUNK]

    const int b   = blockIdx.y;
    const int t0  = blockIdx.x * CHUNK;
    const int tid = threadIdx.x;

    stage_qkv(x, w_qkv, b_qkv, sh_qkv, b, t0, tid);
    __syncthreads();

    // Stage B: v = relu(dwconv(qkv)) channels 32..47 per branch.
    for (int idx = tid; idx < 3 * 16 * CHUNK; idx += 256) {
        int tl = idx & (CHUNK - 1);
        int ch = (idx / CHUNK) & 15;
        int br = idx / (CHUNK * 16);
        const float *w, *bb; int k;
        if      (br == 0) { w = w1; bb = bb1; k = 1; }
        else if (br == 1) { w = w3; bb = bb3; k = 3; }
        else              { w = w5; bb = bb5; k = 5; }
        sh_v[(br * 16 + ch) * CHUNK + tl] = dwconv_one(sh_qkv, w, bb, 32 + ch, tl, k);
    }
    __syncthreads();

    const int wave = tid >> 5;
    const int lane = tid & 31;
    if (wave < 6) {
        const int br   = wave >> 1;
        const int half = wave & 1;
        const int grp  = lane >> 4;
        const int r    = lane & 15;

        // A fragments from qk with block-diagonal head mask (4x4 blocks).
        const float* qkp = qk_ws + ((size_t)(b * 3 + br)) * 256;
        v2f afr[4];
#pragma unroll
        for (int c2 = 0; c2 < 4; ++c2) {
            int e0 = 4 * c2 + 2 * grp;
            float v0 = ((r >> 2) == ( e0      >> 2)) ? qkp[r * 16 + e0]     : 0.0f;
            float v1 = ((r >> 2) == ((e0 + 1) >> 2)) ? qkp[r * 16 + e0 + 1] : 0.0f;
            afr[c2].x = v0; afr[c2].y = v1;
        }

        const int tile0 = half * 4;       // 4 tiles of 16 tokens per half
#pragma unroll
        for (int tile = tile0; tile < tile0 + 4; ++tile) {
            const int tau = tile * 16;
            v8f acc = {};
#pragma unroll
            for (int c2 = 0; c2 < 4; ++c2) {
                int e0 = 4 * c2 + 2 * grp;
                // B[kk][N=r] = V[e0+ (kk-2*grp)][tau + r]
                v2f bm;
                bm.x = sh_v[(br * 16 + e0    ) * CHUNK + tau + r];
                bm.y = sh_v[(br * 16 + e0 + 1) * CHUNK + tau + r];
                acc = __builtin_amdgcn_wmma_f32_16x16x4_f32(
                    false, afr[c2], false, bm, (short)0, acc, false, false);
            }
#pragma unroll
            for (int i = 0; i < 8; ++i) {
                int M = i + 8 * grp;
                sh_out[(br * 16 + M) * CHUNK + tau + r] = acc[i];
            }
        }
    }
    __syncthreads();

    // Stage D: projection y = relu(W_proj * concat + b), coalesced store.
    {
        int tl = tid >> 1;                 // token within chunk (0..127)
        int ob = (tid & 1) * 4;            // output group 0..3 / 4..7
        float* yp = out + (((size_t)b * T_ + t0 + tl) * C_ + ob);
#pragma unroll
        for (int o = 0; o < 4; ++o) {
            float acc = b_proj[ob + o];
#pragma unroll 8
            for (int c = 0; c < TD3; ++c)
                acc += w_proj[(ob + o) * TD3 + c] * sh_out[c * CHUNK + tl];
            yp[o] = fmaxf(acc, 0.0f);
        }
    }
}

// ---------------------------------------------------------------------------
extern "C" void kernel_launch(void* const* d_in, const int* in_sizes, int n_in,
                              void* d_out, int out_size, void* d_ws, size_t ws_size,
                              hipStream_t stream) {
    (void)in_sizes; (void)n_in; (void)out_size; (void)ws_size;
    const float* x      = (const float*)d_in[0];
    const float* w_qkv  = (const float*)d_in[1];
    const float* b_qkv  = (const float*)d_in[2];
    const float* w1     = (const float*)d_in[3];
    const float* bb1    = (const float*)d_in[4];
    const float* w3     = (const float*)d_in[5];
    const float* bb3    = (const float*)d_in[6];
    const float* w5     = (const float*)d_in[7];
    const float* bb5    = (const float*)d_in[8];
    const float* w_proj = (const float*)d_in[9];
    const float* b_proj = (const float*)d_in[10];
    float* qk_ws = (float*)d_ws;                 // B*3*16*16 floats = 196 KB
    float* y     = (float*)d_out;

    const int nqk = B_ * 3 * 256;
    zero_ws_kernel<<<(nqk + 255) / 256, 256, 0, stream>>>(qk_ws, nqk);

    dim3 grid(T_ / CHUNK, B_);
    size_t sh1 = (size_t)(TD3 * QKV_T + 3 * 32 * CHUNK) * sizeof(float);
    litemla_pass1<<<grid, 256, sh1, stream>>>(
        x, w_qkv, b_qkv, w1, bb1, w3, bb3, w5, bb5, qk_ws);

    size_t sh2 = (size_t)(TD3 * QKV_T + 3 * 16 * CHUNK + TD3 * CHUNK) * sizeof(float);
    litemla_pass2<<<grid, 256, sh2, stream>>>(
        x, w_qkv, b_qkv, w1, bb1, w3, bb3, w5, bb5, w_proj, b_proj, qk_ws, y);
}